// SoftDiceLoss_10900626997864
// MI455X (gfx1250) — compile-verified
//
#include <hip/hip_runtime.h>

// Problem constants (from reference): B=64 rows, N=262144 f32 per row.
#define B_    64
#define N_    262144
#define BLKS  64                       // blocks per row for streaming passes
#define TPB   256                      // 8 waves (wave32)
#define EPB   (N_ / BLKS)              // 4096 elements per block
#define F4PB  (EPB / 4)                // 1024 float4 per block

typedef __attribute__((ext_vector_type(2))) float v2f;
typedef __attribute__((ext_vector_type(8))) float v8f;

// ---------------------------------------------------------------------------
// Wave32 full reduction (sum) using V_WMMA_F32_16X16X4_F32.
// A-matrix layout (16x4 f32): lanes 0-15 hold A[m=lane][k=0]=a0, k=1=a1;
// lanes 16-31 hold k=2,k=3. With B = all-ones (layout independent) and a1=0:
//   D[m][n] = q(m) + q(m+16)   for every column n.
// Lane n sums its 8 D VGPRs -> half-sum; shfl_xor(16) completes the wave sum,
// leaving the result broadcast in every lane. Exact for integer-valued floats.
// ---------------------------------------------------------------------------
__device__ __forceinline__ float wave_sum32_wmma(float q) {
  v2f a; a[0] = q;    a[1] = 0.0f;
  v2f b; b[0] = 1.0f; b[1] = 1.0f;
  v8f c = {};
  c = __builtin_amdgcn_wmma_f32_16x16x4_f32(
      /*neg_a=*/false, a, /*neg_b=*/false, b,
      /*c_mod=*/(short)0, c, /*reuse_a=*/false, /*reuse_b=*/false);
  float s = ((c[0] + c[1]) + (c[2] + c[3])) + ((c[4] + c[5]) + (c[6] + c[7]));
  s += __shfl_xor(s, 16, 32);
  return s;
}

__device__ __forceinline__ float wave_max32(float m) {
#pragma unroll
  for (int off = 16; off > 0; off >>= 1)
    m = fmaxf(m, __shfl_xor(m, off, 32));
  return m;
}

// ---- Pass 1: per-block max of targets --------------------------------------
__global__ void k_blockmax(const float* __restrict__ targets,
                           float* __restrict__ bmax) {
  const int row = blockIdx.y, blk = blockIdx.x, tid = threadIdx.x;
  const size_t base = (size_t)row * N_ + (size_t)blk * EPB;
  const float4* t4 = (const float4*)(targets + base);

  float m = -__builtin_inff();
#pragma unroll
  for (int j = 0; j < 4; ++j) {
    float4 t = t4[j * TPB + tid];                  // coalesced global_load_b128
    m = fmaxf(m, fmaxf(fmaxf(t.x, t.y), fmaxf(t.z, t.w)));
  }
  m = wave_max32(m);

  __shared__ float lm[TPB / 32];
  const int wave = tid >> 5, lane = tid & 31;
  if (lane == 0) lm[wave] = m;
  __syncthreads();
  if (tid == 0) {
    float mm = lm[0];
#pragma unroll
    for (int i = 1; i < TPB / 32; ++i) mm = fmaxf(mm, lm[i]);
    bmax[row * BLKS + blk] = mm;
  }
}

// ---- Pass 1b: reduce 64 block maxes -> rowmax[row] -------------------------
__global__ void k_rowmax(const float* __restrict__ bmax,
                         float* __restrict__ rowmax) {
  const int row = blockIdx.x, lane = threadIdx.x;   // 32 threads
  float m = fmaxf(bmax[row * BLKS + lane], bmax[row * BLKS + 32 + lane]);
  m = wave_max32(m);
  if (lane == 0) rowmax[row] = m;
}

// ---- Pass 2: fused {sum_p, sum_t, inter, match_count} per block ------------
__global__ void k_main(const float* __restrict__ probs,
                       const float* __restrict__ targets,
                       const float* __restrict__ rowmax,
                       float4* __restrict__ partials) {
  const int row = blockIdx.y, blk = blockIdx.x, tid = threadIdx.x;
  const size_t base = (size_t)row * N_ + (size_t)blk * EPB;
  const float4* p4 = (const float4*)(probs + base);
  const float4* t4 = (const float4*)(targets + base);

  const float rm = rowmax[row];
  const bool hasPos = rm > 0.0f;

  float sp = 0.0f, st = 0.0f, it = 0.0f;
  int cnt = 0;
#pragma unroll
  for (int j = 0; j < 4; ++j) {
    float4 p = p4[j * TPB + tid];
    float4 t = t4[j * TPB + tid];
#pragma unroll
    for (int k = 0; k < 4; ++k) {
      float pv = (&p.x)[k], tv = (&t.x)[k];
      sp += pv;
      st += tv;
      it = fmaf(pv, tv, it);
      const bool SR = pv > 0.5f;
      const bool GT = hasPos && (tv == rm);
      cnt += (SR == GT) ? 1 : 0;
    }
  }

  // Wave-level reductions via WMMA (EXEC is all-ones: full 256-thread block).
  const float wsp = wave_sum32_wmma(sp);
  const float wst = wave_sum32_wmma(st);
  const float wit = wave_sum32_wmma(it);
  const float wct = wave_sum32_wmma((float)cnt);   // <= 512, exact in f32

  __shared__ float lds[TPB / 32][4];
  const int wave = tid >> 5, lane = tid & 31;
  if (lane == 0) {
    lds[wave][0] = wsp; lds[wave][1] = wst;
    lds[wave][2] = wit; lds[wave][3] = wct;
  }
  __syncthreads();
  if (tid == 0) {
    float a = 0.0f, b = 0.0f, c = 0.0f, d = 0.0f;
#pragma unroll
    for (int i = 0; i < TPB / 32; ++i) {
      a += lds[i][0]; b += lds[i][1]; c += lds[i][2]; d += lds[i][3];
    }
    partials[row * BLKS + blk] = make_float4(a, b, c, d);
  }
}

// ---- Pass 3: reduce 64 block partials -> per-row loss ----------------------
__global__ void k_rowloss(const float4* __restrict__ partials,
                          float* __restrict__ rowloss) {
  const int row = blockIdx.x, lane = threadIdx.x;   // 32 threads, EXEC all-1s
  const float4 a = partials[row * BLKS + lane];
  const float4 b = partials[row * BLKS + 32 + lane];
  float sp = wave_sum32_wmma(a.x + b.x);
  float st = wave_sum32_wmma(a.y + b.y);
  float it = wave_sum32_wmma(a.z + b.z);
  float ct = wave_sum32_wmma(a.w + b.w);           // exact: <= 262144 < 2^24
  if (lane == 0) {
    const float smooth = 1.0f;
    const float dice = (2.0f * it + smooth) / (sp + st + smooth);
    const bool  acc1 = (ct == (float)N_);          // mean(SR==GT) == 1.0
    const float score = acc1 ? 1.0f : dice;
    rowloss[row] = 1.0f - score;
  }
}

// ---- Pass 4: mean over rows -> scalar --------------------------------------
__global__ void k_final(const float* __restrict__ rowloss,
                        float* __restrict__ out) {
  const int lane = threadIdx.x;                     // 32 threads, EXEC all-1s
  float l = rowloss[lane] + rowloss[lane + 32];
  l = wave_sum32_wmma(l);
  if (lane == 0) out[0] = l * (1.0f / (float)B_);
}

extern "C" void kernel_launch(void* const* d_in, const int* in_sizes, int n_in,
                              void* d_out, int out_size, void* d_ws, size_t ws_size,
                              hipStream_t stream) {
  const float* probs   = (const float*)d_in[0];
  const float* targets = (const float*)d_in[1];
  float* out = (float*)d_out;

  // Workspace layout (floats):
  //   [0,      4096)  per-block target maxes
  //   [4096,   4160)  per-row max
  //   [4160,  20544)  per-block float4 partials {sum_p, sum_t, inter, cnt}
  //   [20544, 20608)  per-row loss
  float*  ws       = (float*)d_ws;
  float*  bmax     = ws;
  float*  rowmax   = ws + 4096;
  float4* partials = (float4*)(ws + 4160);
  float*  rowloss  = ws + 20544;

  k_blockmax<<<dim3(BLKS, B_), TPB, 0, stream>>>(targets, bmax);
  k_rowmax  <<<B_, 32, 0, stream>>>(bmax, rowmax);
  k_main    <<<dim3(BLKS, B_), TPB, 0, stream>>>(probs, targets, rowmax, partials);
  k_rowloss <<<B_, 32, 0, stream>>>(partials, rowloss);
  k_final   <<<1,  32, 0, stream>>>(rowloss, out);
}